// singleLGCN_43164421325126
// MI455X (gfx1250) — compile-verified
//
#include <hip/hip_runtime.h>
#include <hip/hip_bf16.h>

// singleLGCN (LightGCN-style 2-layer bipartite GNN), fp32-exact.
// SpMM: COO edge-parallel, wave-per-edge, float4 gather + unsafeAtomicAdd
//       (global_atomic_add_f32). Dense operands are L2-resident (192MB L2).
// GEMM: [M,256]x[256,128] fused concat+bias+ReLU via V_WMMA_F32_16X16X4_F32,
//       A-tile staged in LDS (padded stride), 8 waves/block cover N=128.

#define U_N   200000
#define I_N   100000
#define DD    128
#define NNZ_N 2000000
#define LL    2

typedef __attribute__((ext_vector_type(2))) float v2f;
typedef __attribute__((ext_vector_type(8))) float v8f;

__global__ void zero4_kernel(float4* __restrict__ p, long n4) {
  long i = (long)blockIdx.x * blockDim.x + threadIdx.x;
  if (i < n4) p[i] = make_float4(0.f, 0.f, 0.f, 0.f);
}

// Y[rows[e], :] += vals[e] * X[cols[e], :]   (D = 128, one wave per edge)
__global__ void spmm_atomic_kernel(const int* __restrict__ rows,
                                   const int* __restrict__ cols,
                                   const float* __restrict__ vals,
                                   const float* __restrict__ X,
                                   float* __restrict__ Y, int nnz) {
  int e = blockIdx.x * 8 + (threadIdx.x >> 5);   // 8 waves per 256-thread block
  if (e >= nnz) return;
  int lane = threadIdx.x & 31;
  int r = rows[e];
  int c = cols[e];
  float v = vals[e];
  float4 x4 = ((const float4*)(X + (size_t)c * DD))[lane];  // 32 lanes * 4 = 128
  float* yp = Y + (size_t)r * DD + lane * 4;
  unsafeAtomicAdd(yp + 0, v * x4.x);
  unsafeAtomicAdd(yp + 1, v * x4.y);
  unsafeAtomicAdd(yp + 2, v * x4.z);
  unsafeAtomicAdd(yp + 3, v * x4.w);
}

#define AS_STRIDE 260   // 256 + 4 pad: lane l hits bank (4*l)%64 -> conflict-free, 8B aligned

// Y[M,128] = relu( [HO | X] [M,256] @ W[256,128] + bias ), M % 16 == 0.
// Block: 256 threads = 8 waves; block row-tile = 16 rows; wave w -> N-tile w.
__global__ void gemm_cat_relu_wmma(const float* __restrict__ HO,
                                   const float* __restrict__ X,
                                   const float* __restrict__ W,
                                   const float* __restrict__ bias,
                                   float* __restrict__ Y) {
  __shared__ float As[16 * AS_STRIDE];
  const int mtile = blockIdx.x;
  const int tid   = threadIdx.x;

  // Stage concatenated A tile (16 x 256) into LDS, shared by all 8 waves.
  for (int i = 0; i < 16; ++i) {
    int idx = i * 256 + tid;       // 0..4095
    int row = idx >> 8;            // 0..15
    int k   = idx & 255;           // 0..255
    size_t grow = (size_t)(mtile * 16 + row);
    float v = (k < DD) ? HO[grow * DD + k] : X[grow * DD + (k - DD)];
    As[row * AS_STRIDE + k] = v;
  }
  __syncthreads();

  const int wave  = tid >> 5;          // 0..7 -> N tile
  const int lane  = tid & 31;
  const int mr    = lane & 15;         // A-fragment row (M)
  const int n     = wave * 16 + (lane & 15);   // B/C/D column (N)
  const int khalf = (lane >> 4) << 1;  // 0 for lanes 0-15, 2 for lanes 16-31

  v8f acc = {};
  for (int k0 = 0; k0 < 256; k0 += 4) {
    // A 16x4 fp32 fragment: lane holds {A[mr][k0+khalf], A[mr][k0+khalf+1]}
    v2f a;
    a.x = As[mr * AS_STRIDE + k0 + khalf];
    a.y = As[mr * AS_STRIDE + k0 + khalf + 1];
    // B 4x16 fp32 fragment: lane holds {W[k0+khalf][n], W[k0+khalf+1][n]}
    v2f b;
    b.x = W[(size_t)(k0 + khalf)     * DD + n];
    b.y = W[(size_t)(k0 + khalf + 1) * DD + n];
    acc = __builtin_amdgcn_wmma_f32_16x16x4_f32(false, a, false, b,
                                                (short)0, acc, false, false);
  }

  // C/D layout: VGPR j -> row mtile*16 + j + (lane>=16 ? 8 : 0), col n.
  float bn = bias[n];
  int rbase = mtile * 16 + ((lane >> 4) << 3);
#pragma unroll
  for (int j = 0; j < 8; ++j) {
    float v = acc[j] + bn;
    Y[(size_t)(rbase + j) * DD + n] = v > 0.f ? v : 0.f;
  }
}

extern "C" void kernel_launch(void* const* d_in, const int* in_sizes, int n_in,
                              void* d_out, int out_size, void* d_ws, size_t ws_size,
                              hipStream_t stream) {
  const float* ufea    = (const float*)d_in[0];
  const float* vfea    = (const float*)d_in[1];
  const int*   uv_rows = (const int*)d_in[2];
  const int*   uv_cols = (const int*)d_in[3];
  const float* uv_vals = (const float*)d_in[4];
  const int*   vu_rows = (const int*)d_in[5];
  const int*   vu_cols = (const int*)d_in[6];
  const float* vu_vals = (const float*)d_in[7];
  const float* user_W  = (const float*)d_in[8];   // [L, 256, 128]
  const float* user_b  = (const float*)d_in[9];   // [L, 128]
  const float* item_W  = (const float*)d_in[10];
  const float* item_b  = (const float*)d_in[11];

  const size_t UD = (size_t)U_N * DD;
  const size_t ID = (size_t)I_N * DD;

  // Workspace layout (floats): U0 | I0 | TU(U-sized scratch) | HU | HI
  float* ws = (float*)d_ws;
  float* U0 = ws;
  float* I0 = U0 + UD;
  float* TU = I0 + ID;   // holds tmpI (first I*D), then tmpU (U*D)
  float* HU = TU + UD;
  float* HI = HU + UD;

  float* outU = (float*)d_out;
  float* outI = outU + UD;

  dim3 b256(256);
  const int spmmGrid = (NNZ_N + 7) / 8;

  auto zero = [&](float* p, size_t n) {
    long n4 = (long)(n / 4);
    zero4_kernel<<<dim3((unsigned)((n4 + 255) / 256)), b256, 0, stream>>>((float4*)p, n4);
  };

  const float* cu = ufea;
  const float* ci = vfea;
  for (int l = 0; l < LL; ++l) {
    float* nu = (l == LL - 1) ? outU : U0;
    float* ni = (l == LL - 1) ? outI : I0;

    // user_ho stage 1: tmpI[I,D] = spmm(vu, cu)
    zero(TU, ID);
    spmm_atomic_kernel<<<spmmGrid, b256, 0, stream>>>(vu_rows, vu_cols, vu_vals, cu, TU, NNZ_N);
    // user_ho stage 2: HU[U,D] = spmm(uv, tmpI)
    zero(HU, UD);
    spmm_atomic_kernel<<<spmmGrid, b256, 0, stream>>>(uv_rows, uv_cols, uv_vals, TU, HU, NNZ_N);
    // item_ho stage 1: tmpU[U,D] = spmm(uv, ci)  (reuse TU)
    zero(TU, UD);
    spmm_atomic_kernel<<<spmmGrid, b256, 0, stream>>>(uv_rows, uv_cols, uv_vals, ci, TU, NNZ_N);
    // item_ho stage 2: HI[I,D] = spmm(vu, tmpU)
    zero(HI, ID);
    spmm_atomic_kernel<<<spmmGrid, b256, 0, stream>>>(vu_rows, vu_cols, vu_vals, TU, HI, NNZ_N);

    // Dense fused concat + GEMM + bias + ReLU (exact fp32 WMMA)
    gemm_cat_relu_wmma<<<U_N / 16, b256, 0, stream>>>(
        HU, cu, user_W + (size_t)l * 2 * DD * DD, user_b + (size_t)l * DD, nu);
    gemm_cat_relu_wmma<<<I_N / 16, b256, 0, stream>>>(
        HI, ci, item_W + (size_t)l * 2 * DD * DD, item_b + (size_t)l * DD, ni);

    cu = nu;
    ci = ni;
  }
}